// GCN_60859686584470
// MI455X (gfx1250) — compile-verified
//
#include <hip/hip_runtime.h>
#include <hip/hip_bf16.h>

#define N_NODES 50000
#define N_EDGES 800000
#define FEAT 128
#define EPS_F 1e-5f
#define ROWS_PER_BLK 32

typedef __attribute__((ext_vector_type(2))) float v2f;
typedef __attribute__((ext_vector_type(8))) float v8f;

// ---------------------------------------------------------------------------
// Init: out = bias (broadcast), deg = 1 (self-loop weight), col stats = 0.
// Must run every launch: d_ws/d_out are not re-poisoned between graph replays.
// ---------------------------------------------------------------------------
__global__ void k_init(float* __restrict__ out, const float* __restrict__ b,
                       float* __restrict__ deg, float* __restrict__ colsum,
                       float* __restrict__ colsumsq) {
  long idx = (long)blockIdx.x * blockDim.x + threadIdx.x;
  const long total = (long)N_NODES * FEAT;
  if (idx < total) out[idx] = b[idx & (FEAT - 1)];
  if (idx < N_NODES) deg[idx] = 1.0f;
  if (idx < FEAT) { colsum[idx] = 0.0f; colsumsq[idx] = 0.0f; }
}

// deg[col[e]] += ew[e]
__global__ void k_degree(const int* __restrict__ col, const float* __restrict__ ew,
                         float* __restrict__ deg) {
  int e = blockIdx.x * blockDim.x + threadIdx.x;
  if (e < N_EDGES) atomicAdd(&deg[col[e]], ew[e]);
}

__global__ void k_dinv(const float* __restrict__ deg, float* __restrict__ dinv) {
  int i = blockIdx.x * blockDim.x + threadIdx.x;
  if (i < N_NODES) {
    float d = deg[i];
    dinv[i] = d > 0.0f ? rsqrtf(d) : 0.0f;
  }
}

// ---------------------------------------------------------------------------
// GEMM: h = x @ W  (fp32, exact) via V_WMMA_F32_16X16X4_F32.
// 256 threads = 8 waves; W (64 KB) cached in LDS; wave w computes rows
// [blk*128 + w*16, +16) x all 128 cols (8 accumulator tiles).
// ---------------------------------------------------------------------------
__global__ void __launch_bounds__(256) k_gemm(const float* __restrict__ x,
                                              const float* __restrict__ W,
                                              float* __restrict__ h) {
  __shared__ float Wl[FEAT * FEAT];
  {
    const float4* Wv = (const float4*)W;
    float4* Lv = (float4*)Wl;
    for (int i = threadIdx.x; i < FEAT * FEAT / 4; i += 256) Lv[i] = Wv[i];
  }
  __syncthreads();

  const int wave = threadIdx.x >> 5;
  const int lane = threadIdx.x & 31;
  const int half = lane >> 4;     // 0: K=0,1 / M=0..7 side; 1: K=2,3 / M=8..15 side
  const int l    = lane & 15;
  const int base = blockIdx.x * 128 + wave * 16;

  int arow = base + l;
  if (arow >= N_NODES) arow = N_NODES - 1;   // clamp: keep EXEC all-ones for WMMA
  const float* xr = x + (size_t)arow * FEAT;

  const v8f vzero = {0.f, 0.f, 0.f, 0.f, 0.f, 0.f, 0.f, 0.f};
  v8f acc[8];
#pragma unroll
  for (int i = 0; i < 8; ++i) acc[i] = vzero;

  for (int k0 = 0; k0 < FEAT; k0 += 4) {
    const int kA = k0 + 2 * half;
    v2f a;
    {
      const float2 av = *(const float2*)(xr + kA);   // 8B-aligned (kA even)
      a.x = av.x; a.y = av.y;
    }
#pragma unroll
    for (int nt = 0; nt < 8; ++nt) {
      const int n = nt * 16 + l;
      v2f bf;
      bf.x = Wl[kA * FEAT + n];          // stride-1 across lanes: conflict-free
      bf.y = Wl[(kA + 1) * FEAT + n];
      acc[nt] = __builtin_amdgcn_wmma_f32_16x16x4_f32(
          /*neg_a=*/false, a, /*neg_b=*/false, bf,
          /*c_mod=*/(short)0, acc[nt], /*reuse_a=*/false, /*reuse_b=*/false);
    }
  }

  // C/D layout: lanes 0-15 -> M = v, lanes 16-31 -> M = v+8; N = nt*16 + (lane&15)
#pragma unroll
  for (int nt = 0; nt < 8; ++nt) {
#pragma unroll
    for (int v = 0; v < 8; ++v) {
      const int m = v + (half << 3);
      const int r = base + m;
      if (r < N_NODES) h[(size_t)r * FEAT + nt * 16 + l] = acc[nt][v];
    }
  }
}

// ---------------------------------------------------------------------------
// Edge scatter: out[col] += h[row] * (dinv[row]*ew*dinv[col]).
// h and out are L2-resident (25.6 MB each << 192 MB L2) -> atomics stay on-chip.
// One 128-thread block per edge, one feature per lane.
// ---------------------------------------------------------------------------
__global__ void __launch_bounds__(128)
k_scatter(const int* __restrict__ row, const int* __restrict__ col,
          const float* __restrict__ ew, const float* __restrict__ dinv,
          const float* __restrict__ h, float* __restrict__ out) {
  const int e = blockIdx.x;
  const int f = threadIdx.x;
  const int r = row[e];
  const int c = col[e];
  const float norm = dinv[r] * ew[e] * dinv[c];
  atomicAdd(&out[(size_t)c * FEAT + f], h[(size_t)r * FEAT + f] * norm);
}

// ---------------------------------------------------------------------------
// Self-loop add + ELU + BatchNorm column stats (sum, sumsq) via per-block
// register accumulation + 2 atomics per column per block.
// ---------------------------------------------------------------------------
__global__ void __launch_bounds__(128)
k_elu_stats(float* __restrict__ out, const float* __restrict__ h,
            const float* __restrict__ dinv,
            float* __restrict__ colsum, float* __restrict__ colsumsq) {
  const int f = threadIdx.x;
  const int r0 = blockIdx.x * ROWS_PER_BLK;
  float s = 0.0f, ss = 0.0f;
  for (int i = 0; i < ROWS_PER_BLK; ++i) {
    const int r = r0 + i;
    if (r < N_NODES) {
      const size_t idx = (size_t)r * FEAT + f;
      const float di = dinv[r];
      float v = out[idx] + h[idx] * di * di;     // self-loop: norm = dinv^2
      v = v > 0.0f ? v : expm1f(v);              // ELU (alpha = 1)
      out[idx] = v;
      s += v; ss += v * v;
    }
  }
  atomicAdd(&colsum[f], s);
  atomicAdd(&colsumsq[f], ss);
}

// Fold BN into per-column scale/shift: y = v*scale[f] + shift[f]
__global__ void k_bn_fin(const float* __restrict__ colsum, const float* __restrict__ colsumsq,
                         const float* __restrict__ bn_g, const float* __restrict__ bn_b,
                         float* __restrict__ scale, float* __restrict__ shift) {
  const int f = threadIdx.x;
  const float inv_n = 1.0f / (float)N_NODES;
  const float mu = colsum[f] * inv_n;
  const float var = colsumsq[f] * inv_n - mu * mu;   // biased variance
  const float sc = bn_g[f] * rsqrtf(var + EPS_F);
  scale[f] = sc;
  shift[f] = bn_b[f] - mu * sc;
}

// ---------------------------------------------------------------------------
// LayerNorm per row (in-place on d_out): 128-thread block per row,
// LDS tree reduction (wave32-safe; no 64-lane assumptions).
// ---------------------------------------------------------------------------
__global__ void __launch_bounds__(128)
k_ln(float* __restrict__ out,
     const float* __restrict__ scale, const float* __restrict__ shift,
     const float* __restrict__ ln_g, const float* __restrict__ ln_b) {
  __shared__ float red[FEAT];
  const int r = blockIdx.x;
  const int f = threadIdx.x;
  const size_t idx = (size_t)r * FEAT + f;

  const float v = out[idx] * scale[f] + shift[f];

  red[f] = v;
  __syncthreads();
  for (int s = 64; s > 0; s >>= 1) { if (f < s) red[f] += red[f + s]; __syncthreads(); }
  const float mean = red[0] * (1.0f / FEAT);
  __syncthreads();

  const float d = v - mean;
  red[f] = d * d;
  __syncthreads();
  for (int s = 64; s > 0; s >>= 1) { if (f < s) red[f] += red[f + s]; __syncthreads(); }
  const float var = red[0] * (1.0f / FEAT);

  out[idx] = d * rsqrtf(var + EPS_F) * ln_g[f] + ln_b[f];
}

// ---------------------------------------------------------------------------
extern "C" void kernel_launch(void* const* d_in, const int* in_sizes, int n_in,
                              void* d_out, int out_size, void* d_ws, size_t ws_size,
                              hipStream_t stream) {
  const float* x    = (const float*)d_in[0];
  const int*   eidx = (const int*)d_in[1];     // [2, E] flat: row then col
  const float* ew   = (const float*)d_in[2];
  const float* W    = (const float*)d_in[3];
  const float* b    = (const float*)d_in[4];
  const float* bn_g = (const float*)d_in[5];
  const float* bn_b = (const float*)d_in[6];
  const float* ln_g = (const float*)d_in[7];
  const float* ln_b = (const float*)d_in[8];
  float* out = (float*)d_out;

  float* ws       = (float*)d_ws;
  float* h        = ws;                              // N*128 floats (25.6 MB)
  float* deg      = h + (size_t)N_NODES * FEAT;      // N
  float* dinv     = deg + N_NODES;                   // N
  float* colsum   = dinv + N_NODES;                  // 128
  float* colsumsq = colsum + FEAT;                   // 128
  float* scale    = colsumsq + FEAT;                 // 128
  float* shift    = scale + FEAT;                    // 128

  const int* row = eidx;            // sources
  const int* col = eidx + N_EDGES;  // targets

  const long total = (long)N_NODES * FEAT;
  k_init<<<(unsigned)((total + 255) / 256), 256, 0, stream>>>(out, b, deg, colsum, colsumsq);
  k_degree<<<(N_EDGES + 255) / 256, 256, 0, stream>>>(col, ew, deg);
  k_dinv<<<(N_NODES + 255) / 256, 256, 0, stream>>>(deg, dinv);
  k_gemm<<<(N_NODES + 127) / 128, 256, 0, stream>>>(x, W, h);
  k_scatter<<<N_EDGES, 128, 0, stream>>>(row, col, ew, dinv, h, out);
  k_elu_stats<<<(N_NODES + ROWS_PER_BLK - 1) / ROWS_PER_BLK, 128, 0, stream>>>(out, h, dinv,
                                                                              colsum, colsumsq);
  k_bn_fin<<<1, 128, 0, stream>>>(colsum, colsumsq, bn_g, bn_b, scale, shift);
  k_ln<<<N_NODES, 128, 0, stream>>>(out, scale, shift, ln_g, ln_b);
}